// CRFDecode_19670950216414
// MI455X (gfx1250) — compile-verified
//
#include <hip/hip_runtime.h>
#include <stdint.h>

#define BATCH 1024
#define TT    512
#define CC    128

// ---------- CDNA5 async global->LDS copy (ASYNCcnt path) -------------------
#if __has_builtin(__builtin_amdgcn_global_load_async_to_lds_b32)
#define HAVE_ASYNC_LDS 1
#else
#define HAVE_ASYNC_LDS 0
#endif

typedef __attribute__((address_space(1))) int gbl_i32_t;   // global (AS1)
typedef __attribute__((address_space(3))) int lds_i32_t;   // LDS (AS3)

__device__ __forceinline__ void async_copy_f32(const float* g, float* l) {
#if HAVE_ASYNC_LDS
  __builtin_amdgcn_global_load_async_to_lds_b32(
      (gbl_i32_t*)(uintptr_t)g, (lds_i32_t*)(uint32_t)(uintptr_t)l, 0, 0);
#else
  *l = *g;  // synchronous fallback
#endif
}

__device__ __forceinline__ void async_wait0() {
#if HAVE_ASYNC_LDS
#if __has_builtin(__builtin_amdgcn_s_wait_asynccnt)
  __builtin_amdgcn_s_wait_asynccnt(0);
#else
  asm volatile("s_wait_asynccnt 0" ::: "memory");
#endif
#endif
}

// ---------------------------------------------------------------------------
// One block (128 threads = 4 wave32) per batch element. Thread j owns tag j.
// trans[:,j] in registers; score ping-pong + emissions + history all in LDS.
// ---------------------------------------------------------------------------
extern "C" __global__ __launch_bounds__(CC) void crf_viterbi_kernel(
    const float*   __restrict__ emissions,   // (B,T,C)
    const uint8_t* __restrict__ mask,        // (B,T) bool
    const float*   __restrict__ start_t,     // (C)
    const float*   __restrict__ end_t,       // (C)
    const float*   __restrict__ trans,       // (C,C)
    int*           __restrict__ out)         // (B,T) int32
{
  __shared__ __align__(16) float  sc[2][CC];     // score ping-pong
  __shared__ __align__(16) float  ebuf[2][CC];   // emission double buffer
  __shared__ uint8_t hist[TT][CC];               // backpointers (row 0 unused)
  __shared__ uint8_t mrow[TT];
  __shared__ float   rv[CC];
  __shared__ int     ri[CC];
  __shared__ int     ptag[TT];

  const int j = threadIdx.x;
  const int b = blockIdx.x;
  const float* em = emissions + (size_t)b * TT * CC;

  // Transition column j -> registers (coalesced across lanes, hot in L2).
  float tr[CC];
#pragma unroll
  for (int i = 0; i < CC; ++i) tr[i] = trans[i * CC + j];

  // Mask row -> LDS.
#pragma unroll
  for (int t = j; t < TT; t += CC) mrow[t] = mask[b * TT + t];

  // score0 = start + emissions[:,0]
  float myscore = start_t[j] + em[j];
  sc[0][j] = myscore;

  // Kick off async emission prefetch for t=1.
  async_copy_f32(em + CC + j, &ebuf[1][j]);

  for (int t = 1; t < TT; ++t) {
    async_wait0();        // our wave's emission slice for step t has landed
    __syncthreads();      // prev scores visible; old emission buffer reusable
    if (t + 1 < TT)
      async_copy_f32(em + (size_t)(t + 1) * CC + j, &ebuf[(t + 1) & 1][j]);

    const float* srow = sc[(t - 1) & 1];
    float best = -__builtin_inff();
    int   bidx = 0;
#pragma unroll
    for (int i0 = 0; i0 < CC; i0 += 4) {
      float4 s4 = *reinterpret_cast<const float4*>(&srow[i0]);  // LDS broadcast
      float c0 = s4.x + tr[i0 + 0];
      float c1 = s4.y + tr[i0 + 1];
      float c2 = s4.z + tr[i0 + 2];
      float c3 = s4.w + tr[i0 + 3];
      if (c0 > best) { best = c0; bidx = i0 + 0; }   // strict '>' == first-max
      if (c1 > best) { best = c1; bidx = i0 + 1; }
      if (c2 > best) { best = c2; bidx = i0 + 2; }
      if (c3 > best) { best = c3; bidx = i0 + 3; }
    }
    best += ebuf[t & 1][j];        // emit[j] is constant over i: add after max
    hist[t][j] = (uint8_t)bidx;    // history recorded unconditionally
    if (mrow[t]) myscore = best;   // masked score update
    sc[t & 1][j] = myscore;
  }

  // Final score + end transitions; block argmax with first-max tie-break.
  __syncthreads();
  rv[j] = myscore + end_t[j];
  ri[j] = j;
  __syncthreads();
#pragma unroll
  for (int s = CC / 2; s > 0; s >>= 1) {
    if (j < s) {
      float ov = rv[j + s]; int oi = ri[j + s];
      if (ov > rv[j] || (ov == rv[j] && oi < ri[j])) { rv[j] = ov; ri[j] = oi; }
    }
    __syncthreads();
  }

  // Backtrack entirely in LDS (thread 0): 511 dependent u8 loads, on-chip.
  if (j == 0) {
    int tag = ri[0];
    ptag[TT - 1] = tag;
    for (int t = TT - 1; t >= 1; --t) {
      int prev = hist[t][tag];
      if (mrow[t]) tag = prev;
      ptag[t - 1] = tag;
    }
  }
  __syncthreads();

  // Coalesced path writeout.
#pragma unroll
  for (int t = j; t < TT; t += CC) out[(size_t)b * TT + t] = ptag[t];
}

extern "C" void kernel_launch(void* const* d_in, const int* in_sizes, int n_in,
                              void* d_out, int out_size, void* d_ws, size_t ws_size,
                              hipStream_t stream) {
  const float*   emissions = (const float*)  d_in[0];
  const uint8_t* mask      = (const uint8_t*)d_in[1];
  const float*   start_t   = (const float*)  d_in[2];
  const float*   end_t     = (const float*)  d_in[3];
  const float*   trans     = (const float*)  d_in[4];
  int*           out       = (int*)d_out;
  (void)in_sizes; (void)n_in; (void)out_size; (void)d_ws; (void)ws_size;

  crf_viterbi_kernel<<<BATCH, CC, 0, stream>>>(emissions, mask, start_t,
                                               end_t, trans, out);
}